// ICNN_17884243820732
// MI455X (gfx1250) — compile-verified
//
#include <hip/hip_runtime.h>
#include <hip/hip_bf16.h>

// MI455X (gfx1250) wave32 ICNN forward+backward.
// Per wave: 16 batch elements; activations held in WMMA f32 C/D layout
// (hidden index -> VGPR slot/lane-half, element -> lane&15).
// Four 32x32 matmuls per group done with V_WMMA_F32_16X16X4_F32.
// Softplus+sigmoid share one exp; raw v_exp/v_log used (args provably in
// safe range: exp arg <= 0, log arg in (1,2]) to skip libm range fixups.
// grad_ref computed once by a tiny prologue kernel into d_ws.

typedef __attribute__((ext_vector_type(2))) float v2f;
typedef __attribute__((ext_vector_type(8))) float v8f;

#define DINL __device__ __forceinline__

#if defined(__HIP_DEVICE_COMPILE__) && __has_builtin(__builtin_amdgcn_exp2f)
#define EXP2F(x) __builtin_amdgcn_exp2f(x)
#else
#define EXP2F(x) exp2f(x)
#endif
#if defined(__HIP_DEVICE_COMPILE__) && __has_builtin(__builtin_amdgcn_logf)
#define LOG2F(x) __builtin_amdgcn_logf(x)
#else
#define LOG2F(x) log2f(x)
#endif
#if defined(__HIP_DEVICE_COMPILE__) && __has_builtin(__builtin_amdgcn_rcpf)
#define RCPF(x) __builtin_amdgcn_rcpf(x)
#else
#define RCPF(x) (1.0f / (x))
#endif

constexpr float kActScale = 1.0f / 12.0f;
constexpr float kLog2e = 1.44269504088896340736f;
constexpr float kLn2 = 0.69314718055994530942f;

// t = e^{-|x|} via raw v_exp (arg <= 0: no overflow possible)
DINL float exp_negabs(float x) {
    return EXP2F(-fabsf(x) * kLog2e);
}

DINL float softplusf(float x) {
    // stable: max(x,0) + log1p(exp(-|x|)); log arg in (1,2] -> raw v_log safe
    float t = exp_negabs(x);
    return fmaxf(x, 0.f) + kLn2 * LOG2F(1.f + t);
}

// Fused softplus + sigmoid sharing a single exp:
//   t = e^{-|x|};  sp = max(x,0) + log(1+t);  sig = x>=0 ? 1/(1+t) : t/(1+t)
DINL void sp_sig(float x, float& sp, float& sg) {
    float t = exp_negabs(x);
    float u = 1.f + t;  // in (1, 2]
    float r = RCPF(u);
    sp = fmaxf(x, 0.f) + kLn2 * LOG2F(u);
    sg = (x >= 0.f) ? r : t * r;
}

DINL float sig_only(float x) {
    float t = exp_negabs(x);
    float r = RCPF(1.f + t);
    return (x >= 0.f) ? r : t * r;
}

// Swap lane halves (lane i <-> lane i^16) via DS_SWIZZLE_B32, SWAPX16 pattern:
// offset[15]=0 (group-of-32), xor=0x10, and=0x1f -> 0x401F.
DINL float swap_half(float x) {
    return __int_as_float(__builtin_amdgcn_ds_swizzle(__float_as_int(x), 0x401f));
}

// Build B operand (4x16, f32) for K-step S from a 32x16 tile stored in
// C/D layout: x[t*8+j] holds h=t*16+j (lanes 0-15) and h=t*16+j+8 (lanes 16-31).
// B layout: vgpr q: lanes 0-15 hold row K=4S+q, lanes 16-31 hold row K=4S+2+q.
template <int S>
DINL v2f buildB(const float (&x)[16], bool lo) {
    constexpr int h0 = 4 * S + 0, h1 = 4 * S + 1, h2 = 4 * S + 2, h3 = 4 * S + 3;
    constexpr int i0 = ((h0 >> 4) << 3) | (h0 & 7);
    constexpr int i1 = ((h1 >> 4) << 3) | (h1 & 7);
    constexpr int i2 = ((h2 >> 4) << 3) | (h2 & 7);
    constexpr int i3 = ((h3 >> 4) << 3) | (h3 & 7);
    constexpr bool s0 = ((h0 & 15) >= 8);  // stored in hi lanes?
    constexpr bool s1 = ((h1 & 15) >= 8);
    constexpr bool s2 = ((h2 & 15) >= 8);
    constexpr bool s3 = ((h3 & 15) >= 8);
    float a0 = s0 ? swap_half(x[i0]) : x[i0];   // h0 value into lanes 0-15
    float a1 = s1 ? swap_half(x[i1]) : x[i1];
    float b0 = s2 ? x[i2] : swap_half(x[i2]);   // h2 value into lanes 16-31
    float b1 = s3 ? x[i3] : swap_half(x[i3]);
    v2f r;
    r.x = lo ? a0 : b0;
    r.y = lo ? a1 : b1;
    return r;
}

// One K-step (K=4S..4S+3) of Vt(32x16) += W(32x32) * Xt(32x16).
// A layout: lane m=lane&15 holds row (mtile*16+m); vgpr0/1 = K=4S+2*khalf, +1.
template <int S>
DINL void kstep(const float* W, const float (&x)[16], v8f& c0, v8f& c1,
                int row, int khalf, bool lo) {
    v2f b = buildB<S>(x, lo);
    int base = row * 32 + 4 * S + 2 * khalf;
    v2f a0 = *(const v2f*)(W + base);         // rows 0..15
    v2f a1 = *(const v2f*)(W + base + 512);   // rows 16..31
    c0 = __builtin_amdgcn_wmma_f32_16x16x4_f32(false, a0, false, b, (short)0, c0,
                                               false, false);
    c1 = __builtin_amdgcn_wmma_f32_16x16x4_f32(false, a1, false, b, (short)0, c1,
                                               false, false);
}

DINL void matmul32(const float* W, const float (&x)[16], float (&y)[16],
                   int row, int khalf, bool lo) {
    v8f c0 = {0.f, 0.f, 0.f, 0.f, 0.f, 0.f, 0.f, 0.f};
    v8f c1 = {0.f, 0.f, 0.f, 0.f, 0.f, 0.f, 0.f, 0.f};
    kstep<0>(W, x, c0, c1, row, khalf, lo);
    kstep<1>(W, x, c0, c1, row, khalf, lo);
    kstep<2>(W, x, c0, c1, row, khalf, lo);
    kstep<3>(W, x, c0, c1, row, khalf, lo);
    kstep<4>(W, x, c0, c1, row, khalf, lo);
    kstep<5>(W, x, c0, c1, row, khalf, lo);
    kstep<6>(W, x, c0, c1, row, khalf, lo);
    kstep<7>(W, x, c0, c1, row, khalf, lo);
#pragma unroll
    for (int j = 0; j < 8; ++j) {
        y[j] = c0[j];
        y[8 + j] = c1[j];
    }
}

// Full fwd+bwd for 16 elements; returns per-lane partial d/dI1, d/dI2
// (must be half-summed by caller). Constant Bo_w term skipped (cancels vs ref).
DINL void process_group(float I1, float I2,
                        const float* sW1, const float* sW1T,
                        const float* sW2, const float* sW2T,
                        const float4* sC0, const float4* sC1, const float4* sC2,
                        int row, int khalf, bool lo,
                        float& accI1, float& accI2) {
    float x[16], v[16], spu1[16], sigu0[16], sigu1[16], d1[16];
    accI1 = 0.f;
    accI2 = 0.f;
    // ---- forward: z1 = sp(u0); also u1/u2 branches; fold u2 direct-grad path
#pragma unroll
    for (int t = 0; t < 2; ++t) {
#pragma unroll
        for (int j = 0; j < 8; ++j) {
            int slot = t * 8 + j;
            int idx = t * 16 + j + khalf * 8;  // hidden index h for this lane
            float4 cf0 = sC0[idx];
            float u0 = fmaf(cf0.x, I1, fmaf(cf0.y, I2, cf0.z));
            sp_sig(u0, x[slot], sigu0[slot]);
            float4 cf1 = sC1[idx];
            float u1 = fmaf(cf1.x, I1, fmaf(cf1.y, I2, cf1.z));
            sp_sig(u1, spu1[slot], sigu1[slot]);
            float4 cf2 = sC2[idx];
            float u2 = fmaf(cf2.x, I1, fmaf(cf2.y, I2, cf2.z));
            float s2 = cf0.w * sig_only(u2);   // w_out[h] * sigma(u2)
            accI1 = fmaf(s2, cf2.x, accI1);    // @ B2_w
            accI2 = fmaf(s2, cf2.y, accI2);
        }
    }
    matmul32(sW1, x, v, row, khalf, lo);  // v1 = sp(A1) . z1
#pragma unroll
    for (int s = 0; s < 16; ++s) {
        float spv, sg;
        sp_sig(v[s], spv, sg);
        x[s] = fmaf(kActScale * spv, spv, spu1[s]);  // z2
        d1[s] = 2.f * kActScale * spv * sg;          // dz2/dv1 factor
    }
    matmul32(sW2, x, v, row, khalf, lo);  // v2 = sp(A2) . z2
    // ---- backward
#pragma unroll
    for (int t = 0; t < 2; ++t) {
#pragma unroll
        for (int j = 0; j < 8; ++j) {
            int slot = t * 8 + j;
            int idx = t * 16 + j + khalf * 8;
            float wout = sC0[idx].w;
            float spv, sg;
            sp_sig(v[slot], spv, sg);
            x[slot] = wout * 2.f * kActScale * spv * sg;  // gv2
        }
    }
    matmul32(sW2T, x, v, row, khalf, lo);  // gz2 = sp(A2)^T . gv2
#pragma unroll
    for (int t = 0; t < 2; ++t) {
#pragma unroll
        for (int j = 0; j < 8; ++j) {
            int slot = t * 8 + j;
            int idx = t * 16 + j + khalf * 8;
            float4 cf1 = sC1[idx];
            float tmp = v[slot] * sigu1[slot];
            accI1 = fmaf(tmp, cf1.x, accI1);  // @ B1_w
            accI2 = fmaf(tmp, cf1.y, accI2);
            x[slot] = v[slot] * d1[slot];     // gv1
        }
    }
    matmul32(sW1T, x, v, row, khalf, lo);  // gz1 = sp(A1)^T . gv1
#pragma unroll
    for (int t = 0; t < 2; ++t) {
#pragma unroll
        for (int j = 0; j < 8; ++j) {
            int slot = t * 8 + j;
            int idx = t * 16 + j + khalf * 8;
            float4 cf0 = sC0[idx];
            float tmp = v[slot] * sigu0[slot];
            accI1 = fmaf(tmp, cf0.x, accI1);  // @ Bf_w
            accI2 = fmaf(tmp, cf0.y, accI2);
        }
    }
}

// Shared LDS preprocessing: softplus'd weight matrices (+transposes) and
// packed per-hidden coefficient float4s.
DINL void prep_lds(int tid, int nthreads,
                   const float* Bf_w, const float* Bf_b, const float* A1,
                   const float* B1_w, const float* B1_b, const float* A2,
                   const float* B2_w, const float* B2_b, const float* A_out,
                   float* sW1, float* sW1T, float* sW2, float* sW2T,
                   float4* sC0, float4* sC1, float4* sC2) {
    for (int i = tid; i < 1024; i += nthreads) {
        int r = i >> 5, c = i & 31;
        sW1[i] = softplusf(A1[i]);
        sW1T[i] = softplusf(A1[c * 32 + r]);
        sW2[i] = softplusf(A2[i]);
        sW2T[i] = softplusf(A2[c * 32 + r]);
    }
    for (int i = tid; i < 32; i += nthreads) {
        sC0[i] = make_float4(Bf_w[2 * i], Bf_w[2 * i + 1], Bf_b[i],
                             softplusf(A_out[i]));
        sC1[i] = make_float4(B1_w[2 * i], B1_w[2 * i + 1], B1_b[i], 0.f);
        sC2[i] = make_float4(B2_w[2 * i], B2_w[2 * i + 1], B2_b[i], 0.f);
    }
    __syncthreads();
}

// One wave: compute grad_ref = dW/dz0 at z0 = 0, store {refI1, refI2} to ws.
__global__ __launch_bounds__(32) void icnn_ref_kernel(
    const float* __restrict__ Bf_w, const float* __restrict__ Bf_b,
    const float* __restrict__ A1, const float* __restrict__ B1_w,
    const float* __restrict__ B1_b, const float* __restrict__ A2,
    const float* __restrict__ B2_w, const float* __restrict__ B2_b,
    const float* __restrict__ A_out, float* __restrict__ ws) {
    __shared__ float sW1[1024], sW1T[1024], sW2[1024], sW2T[1024];
    __shared__ float4 sC0[32], sC1[32], sC2[32];
    int tid = threadIdx.x;
    prep_lds(tid, 32, Bf_w, Bf_b, A1, B1_w, B1_b, A2, B2_w, B2_b, A_out,
             sW1, sW1T, sW2, sW2T, sC0, sC1, sC2);
    int lane = tid & 31;
    int row = lane & 15;
    int khalf = lane >> 4;
    bool lo = (khalf == 0);
    float rI1, rI2;
    process_group(0.f, 0.f, sW1, sW1T, sW2, sW2T, sC0, sC1, sC2, row, khalf, lo,
                  rI1, rI2);
    float refI1 = rI1 + swap_half(rI1);
    float refI2 = rI2 + swap_half(rI2);
    if (tid == 0) {
        ws[0] = refI1;
        ws[1] = refI2;
    }
}

__global__ __launch_bounds__(256) void icnn_kernel(
    const float* __restrict__ eps, const float* __restrict__ Bf_w,
    const float* __restrict__ Bf_b, const float* __restrict__ A1,
    const float* __restrict__ B1_w, const float* __restrict__ B1_b,
    const float* __restrict__ A2, const float* __restrict__ B2_w,
    const float* __restrict__ B2_b, const float* __restrict__ A_out,
    const float* __restrict__ ws, float* __restrict__ out, int n,
    int total_waves) {
    __shared__ float sW1[1024], sW1T[1024], sW2[1024], sW2T[1024];
    __shared__ float4 sC0[32], sC1[32], sC2[32];

    int tid = threadIdx.x;
    prep_lds(tid, 256, Bf_w, Bf_b, A1, B1_w, B1_b, A2, B2_w, B2_b, A_out,
             sW1, sW1T, sW2, sW2T, sC0, sC1, sC2);

    int lane = tid & 31;
    int row = lane & 15;
    int khalf = lane >> 4;
    bool lo = (khalf == 0);
    int wave = blockIdx.x * (blockDim.x >> 5) + (tid >> 5);

    float refI1 = ws[0];  // grad_ref from prologue kernel
    float refI2 = ws[1];

    int ngroups = (n + 15) >> 4;
    for (int g = wave; g < ngroups; g += total_waves) {
        int e = g * 16 + row;
        int ec = e < n ? e : (n - 1);  // clamp: keep EXEC all-ones for WMMA
        float e11 = eps[ec * 3 + 0];
        float e22 = eps[ec * 3 + 1];
        float e12 = eps[ec * 3 + 2];
        float I1 = e11 + e22;
        float I2 = e11 * e11 + 2.f * e12 * e12 + e22 * e22;
        float aI1, aI2;
        process_group(I1, I2, sW1, sW1T, sW2, sW2T, sC0, sC1, sC2, row, khalf,
                      lo, aI1, aI2);
        float gI1 = aI1 + swap_half(aI1) - refI1;
        float gI2 = aI2 + swap_half(aI2) - refI2;
        if (lo && e < n) {
            out[e * 3 + 0] = gI1 + 2.f * e11 * gI2;  // s11
            out[e * 3 + 1] = gI1 + 2.f * e22 * gI2;  // s22
            out[e * 3 + 2] = 2.f * e12 * gI2;        // s12
        }
    }
}

extern "C" void kernel_launch(void* const* d_in, const int* in_sizes, int n_in,
                              void* d_out, int out_size, void* d_ws,
                              size_t ws_size, hipStream_t stream) {
    const float* eps = (const float*)d_in[0];
    const float* Bf_w = (const float*)d_in[1];
    const float* Bf_b = (const float*)d_in[2];
    const float* A1 = (const float*)d_in[3];
    const float* B1_w = (const float*)d_in[4];
    const float* B1_b = (const float*)d_in[5];
    const float* A2 = (const float*)d_in[6];
    const float* B2_w = (const float*)d_in[7];
    const float* B2_b = (const float*)d_in[8];
    const float* A_out = (const float*)d_in[9];
    // d_in[10]=Bo_w, d_in[11]=Bo_b: constant terms cancel against grad_ref.
    float* out = (float*)d_out;
    float* ws = (float*)d_ws;

    int n = in_sizes[0] / 3;  // batch elements
    const int threads = 256;  // 8 wave32s per block
    const int blocks = 1024;
    int total_waves = blocks * (threads / 32);

    icnn_ref_kernel<<<1, 32, 0, stream>>>(Bf_w, Bf_b, A1, B1_w, B1_b, A2, B2_w,
                                          B2_b, A_out, ws);
    icnn_kernel<<<blocks, threads, 0, stream>>>(eps, Bf_w, Bf_b, A1, B1_w, B1_b,
                                                A2, B2_w, B2_b, A_out, ws, out,
                                                n, total_waves);
}